// SymplecticODEEncoder_54640573940236
// MI455X (gfx1250) — compile-verified
//
#include <hip/hip_runtime.h>

typedef float v2f __attribute__((ext_vector_type(2)));
typedef float v8f __attribute__((ext_vector_type(8)));

#define B_  32
#define T_  256
#define D_  128
#define Z_  64
#define H_  256

__device__ __forceinline__ v8f wmma4(v2f a, v2f b, v8f c) {
  // D = A(16x4 f32) * B(4x16 f32) + C(16x16 f32)
  return __builtin_amdgcn_wmma_f32_16x16x4_f32(false, a, false, b, (short)0, c,
                                               false, false);
}

// ---------------------------------------------------------------------------
// Kernel A: two-layer tanh RNN over reversed time. One workgroup (8 waves),
// weights + hidden states in LDS. rnn_out[b][tt] is written to out[b][tt-1]
// (the z0 the ODE kernel integrates in place); rnn_out[b][T-1] also goes to
// out[b][T-1] (passthrough last column of the output).
// ---------------------------------------------------------------------------
__global__ __launch_bounds__(256) void rnn_kernel(
    const float* __restrict__ x, const float* __restrict__ h0,
    const float* __restrict__ Wih0, const float* __restrict__ Whh0,
    const float* __restrict__ bih0, const float* __restrict__ bhh0,
    const float* __restrict__ Wih1, const float* __restrict__ Whh1,
    const float* __restrict__ bih1, const float* __restrict__ bhh1,
    float* __restrict__ out) {
  extern __shared__ float sm[];
  const int WIH0 = 0;            // 64*128
  const int WHH0 = 8192;         // 64*64
  const int WIH1 = 12288;        // 64*64
  const int WHH1 = 16384;        // 64*64
  const int HB0  = 20480;        // 32 rows * 68 stride
  const int HB1  = 22656;        // 32 rows * 68 stride

  const int tid = threadIdx.x;
  for (int i = tid; i < Z_ * D_; i += 256) sm[WIH0 + i] = Wih0[i];
  for (int i = tid; i < Z_ * Z_; i += 256) sm[WHH0 + i] = Whh0[i];
  for (int i = tid; i < Z_ * Z_; i += 256) sm[WIH1 + i] = Wih1[i];
  for (int i = tid; i < Z_ * Z_; i += 256) sm[WHH1 + i] = Whh1[i];
  for (int i = tid; i < B_ * Z_; i += 256) {
    int r = i >> 6, c = i & 63;
    sm[HB0 + r * 68 + c] = h0[c];        // layer-0 initial state
    sm[HB1 + r * 68 + c] = h0[Z_ + c];   // layer-1 initial state
  }
  __syncthreads();

  const int lane   = tid & 31;
  const int w      = tid >> 5;
  const int mt     = w >> 2;               // batch tile (0..1)
  const int nt     = w & 3;                // z tile (0..3)
  const int colT   = lane & 15;            // C column / A row within tile
  const int koff   = (lane >> 4) << 1;     // K sub-offset (0 or 2)
  const int rowOff = (lane >> 4) << 3;     // C row offset (0 or 8)

  const int col   = nt * 16 + colT;        // z index owned by this lane
  const int arow  = mt * 16 + colT;        // batch row fed as A operand
  const float bias0 = bih0[col] + bhh0[col];
  const float bias1 = bih1[col] + bhh1[col];

  for (int s = 0; s < T_; ++s) {
    const int tt = T_ - 1 - s;             // reversed time index into x
    // ---------- layer 0: tanh(x_t @ Wih0^T + h @ Whh0^T + b) ----------
    v8f c0 = {};
    const float* xrow = x + ((size_t)arow * T_ + tt) * D_;
#pragma unroll 4
    for (int kb = 0; kb < D_; kb += 4) {
      v2f a = *(const v2f*)(xrow + kb + koff);
      v2f b = *(const v2f*)(&sm[WIH0 + col * D_ + kb + koff]);
      c0 = wmma4(a, b, c0);
    }
#pragma unroll 4
    for (int kb = 0; kb < Z_; kb += 4) {
      v2f a = *(const v2f*)(&sm[HB0 + arow * 68 + kb + koff]);
      v2f b = *(const v2f*)(&sm[WHH0 + col * Z_ + kb + koff]);
      c0 = wmma4(a, b, c0);
    }
    float y0[8];
#pragma unroll
    for (int r = 0; r < 8; ++r) y0[r] = tanhf(c0[r] + bias0);
    __syncthreads();
#pragma unroll
    for (int r = 0; r < 8; ++r)
      sm[HB0 + (mt * 16 + rowOff + r) * 68 + col] = y0[r];
    __syncthreads();

    // ---------- layer 1 ----------
    v8f c1 = {};
#pragma unroll 4
    for (int kb = 0; kb < Z_; kb += 4) {
      v2f a = *(const v2f*)(&sm[HB0 + arow * 68 + kb + koff]);
      v2f b = *(const v2f*)(&sm[WIH1 + col * Z_ + kb + koff]);
      c1 = wmma4(a, b, c1);
    }
#pragma unroll 4
    for (int kb = 0; kb < Z_; kb += 4) {
      v2f a = *(const v2f*)(&sm[HB1 + arow * 68 + kb + koff]);
      v2f b = *(const v2f*)(&sm[WHH1 + col * Z_ + kb + koff]);
      c1 = wmma4(a, b, c1);
    }
    float y1[8];
#pragma unroll
    for (int r = 0; r < 8; ++r) y1[r] = tanhf(c1[r] + bias1);
    __syncthreads();
#pragma unroll
    for (int r = 0; r < 8; ++r) {
      int brow = mt * 16 + rowOff + r;
      sm[HB1 + brow * 68 + col] = y1[r];
      if (tt >= 1)
        out[((size_t)brow * T_ + (tt - 1)) * Z_ + col] = y1[r];
      if (tt == T_ - 1)
        out[((size_t)brow * T_ + (T_ - 1)) * Z_ + col] = y1[r];
    }
    __syncthreads();
  }
}

// ---------------------------------------------------------------------------
// One-time weight transpose into workspace so the backward-pass WMMA
// B-operands become contiguous float2 loads (global_load_b64) instead of
// 1KB-strided scalar pairs. ws layout: Wh1T[256*256] then Wh0T[64*256].
// ---------------------------------------------------------------------------
__global__ __launch_bounds__(256) void transpose_kernel(
    const float* __restrict__ Wh1, const float* __restrict__ Wh0,
    float* __restrict__ ws) {
  int i = blockIdx.x * 256 + threadIdx.x;
  if (i < H_ * H_) {
    int n = i >> 8, k = i & 255;
    ws[n * H_ + k] = Wh1[k * H_ + n];
  }
  if (i < Z_ * H_) {
    int n = i >> 8, k = i & 255;
    ws[H_ * H_ + n * H_ + k] = Wh0[k * Z_ + n];
  }
}

// ---------------------------------------------------------------------------
// Kernel B: Yoshida-4 symplectic integration of 8160 independent z rows.
// 255 blocks x 32 rows. z + activations in LDS; weights stream from L2.
// TR=true: backward B-operands come from transposed copies in ws.
// ---------------------------------------------------------------------------
template <bool TR>
__global__ __launch_bounds__(256) void ode_kernel(
    const float* __restrict__ Wh0, const float* __restrict__ bh0,
    const float* __restrict__ Wh1, const float* __restrict__ bh1,
    const float* __restrict__ Wout,
    const float* __restrict__ Wh1T,   // [n*256+k] if TR, else == Wh1
    const float* __restrict__ Wh0T,   // [n*256+k] if TR, else == Wh0
    float* __restrict__ out) {
  extern __shared__ float sm[];
  const int ZB   = 0;                 // 32 x 68
  const int H1B  = 2176;              // 32 x 264
  const int DA1B = H1B + 32 * 264;
  const int DA0B = DA1B + 32 * 264;

  const int tid    = threadIdx.x;
  const int lane   = tid & 31;
  const int w      = tid >> 5;
  const int colT   = lane & 15;
  const int koff   = (lane >> 4) << 1;
  const int rowOff = (lane >> 4) << 3;

  // Load this block's 32-row z tile (in-place buffer in d_out).
  for (int i = tid; i < 32 * Z_; i += 256) {
    int r = i >> 6, c = i & 63;
    int gr = blockIdx.x * 32 + r;
    int b = gr / (T_ - 1), t = gr % (T_ - 1);
    sm[ZB + r * 68 + c] = out[((size_t)b * T_ + t) * Z_ + c];
  }
  __syncthreads();

  const float hstep = 0.02f;                       // DT / DISC
  const float w1c = 1.3512071919596578f;           // 1/(2-cbrt(2))
  const float w0c = -1.7024143839193153f;          // -cbrt(2)*w1
  const float qc[4] = {0.5f * w1c * hstep, 0.5f * (w0c + w1c) * hstep,
                       0.5f * (w0c + w1c) * hstep, 0.5f * w1c * hstep};
  const float pc[3] = {-w1c * hstep, -w0c * hstep, -w1c * hstep};

  for (int step = 0; step < 4; ++step) {           // DISC-1 steps
    for (int sub = 0; sub < 7; ++sub) {            // drift,kick,...,drift
      const bool drift = (sub & 1) == 0;
      const float coef = drift ? qc[sub >> 1] : pc[sub >> 1];

      // phase 1: h1 = tanh(z @ Wh0^T + bh0)   (M=32, N=256, K=64)
      for (int nn = 0; nn < 2; ++nn) {
        const int n = (w + nn * 8) * 16 + colT;
        __builtin_prefetch(Wh0 + (size_t)n * Z_, 0, 1);
        v8f c0 = {}, c1 = {};
#pragma unroll 4
        for (int kb = 0; kb < Z_; kb += 4) {
          int k = kb + koff;
          v2f b  = *(const v2f*)(Wh0 + (size_t)n * Z_ + k);
          v2f a0 = *(const v2f*)(&sm[ZB + colT * 68 + k]);
          v2f a1 = *(const v2f*)(&sm[ZB + (16 + colT) * 68 + k]);
          c0 = wmma4(a0, b, c0);
          c1 = wmma4(a1, b, c1);
        }
        const float bias = bh0[n];
#pragma unroll
        for (int r = 0; r < 8; ++r) {
          sm[H1B + (rowOff + r) * 264 + n]      = tanhf(c0[r] + bias);
          sm[H1B + (16 + rowOff + r) * 264 + n] = tanhf(c1[r] + bias);
        }
      }
      __syncthreads();

      // phase 2: da1 = Wout * (1 - tanh(h1 @ Wh1^T + bh1)^2)  (K=256)
      for (int nn = 0; nn < 2; ++nn) {
        const int n = (w + nn * 8) * 16 + colT;
        __builtin_prefetch(Wh1 + (size_t)n * H_, 0, 1);
        v8f c0 = {}, c1 = {};
#pragma unroll 4
        for (int kb = 0; kb < H_; kb += 4) {
          int k = kb + koff;
          v2f b  = *(const v2f*)(Wh1 + (size_t)n * H_ + k);
          v2f a0 = *(const v2f*)(&sm[H1B + colT * 264 + k]);
          v2f a1 = *(const v2f*)(&sm[H1B + (16 + colT) * 264 + k]);
          c0 = wmma4(a0, b, c0);
          c1 = wmma4(a1, b, c1);
        }
        const float bias = bh1[n];
        const float wo   = Wout[n];
#pragma unroll
        for (int r = 0; r < 8; ++r) {
          float t0 = tanhf(c0[r] + bias);
          float t1 = tanhf(c1[r] + bias);
          sm[DA1B + (rowOff + r) * 264 + n]      = wo * (1.0f - t0 * t0);
          sm[DA1B + (16 + rowOff + r) * 264 + n] = wo * (1.0f - t1 * t1);
        }
      }
      __syncthreads();

      // phase 3: da0 = (da1 @ Wh1) * (1 - h1^2)   (K=256)
      for (int nn = 0; nn < 2; ++nn) {
        const int n = (w + nn * 8) * 16 + colT;
        if (TR) __builtin_prefetch(Wh1T + (size_t)n * H_, 0, 1);
        v8f c0 = {}, c1 = {};
#pragma unroll 4
        for (int kb = 0; kb < H_; kb += 4) {
          int k = kb + koff;
          v2f b;
          if (TR) {
            b = *(const v2f*)(Wh1T + (size_t)n * H_ + k);     // contiguous
          } else {
            b.x = Wh1T[(size_t)k * H_ + n];                    // strided
            b.y = Wh1T[(size_t)(k + 1) * H_ + n];
          }
          v2f a0 = *(const v2f*)(&sm[DA1B + colT * 264 + k]);
          v2f a1 = *(const v2f*)(&sm[DA1B + (16 + colT) * 264 + k]);
          c0 = wmma4(a0, b, c0);
          c1 = wmma4(a1, b, c1);
        }
#pragma unroll
        for (int r = 0; r < 8; ++r) {
          int r0 = rowOff + r, r1 = 16 + rowOff + r;
          float ha = sm[H1B + r0 * 264 + n];
          float hb = sm[H1B + r1 * 264 + n];
          sm[DA0B + r0 * 264 + n] = c0[r] * (1.0f - ha * ha);
          sm[DA0B + r1 * 264 + n] = c1[r] * (1.0f - hb * hb);
        }
      }
      __syncthreads();

      // phase 4: dz = da0 @ Wh0 (M=32, N=64, K=256); apply drift/kick.
      {
        const int mt = w >> 2;
        const int n  = (w & 3) * 16 + colT;   // z column 0..63
        if (TR) __builtin_prefetch(Wh0T + (size_t)n * H_, 0, 1);
        v8f cc = {};
#pragma unroll 4
        for (int kb = 0; kb < H_; kb += 4) {
          int k = kb + koff;
          v2f b;
          if (TR) {
            b = *(const v2f*)(Wh0T + (size_t)n * H_ + k);     // contiguous
          } else {
            b.x = Wh0T[(size_t)k * Z_ + n];                    // strided
            b.y = Wh0T[(size_t)(k + 1) * Z_ + n];
          }
          v2f a = *(const v2f*)(&sm[DA0B + (mt * 16 + colT) * 264 + k]);
          cc = wmma4(a, b, cc);
        }
#pragma unroll
        for (int r = 0; r < 8; ++r) {
          int row = mt * 16 + rowOff + r;
          float dz = cc[r];
          if (drift) {            // q += cs*h * dH/dp  (p-grad cols 32..63)
            if (n >= 32) sm[ZB + row * 68 + (n - 32)] += coef * dz;
          } else {                // p -= ds*h * dH/dq  (q-grad cols 0..31)
            if (n < 32)  sm[ZB + row * 68 + (n + 32)] += coef * dz;
          }
        }
      }
      __syncthreads();
    }
  }

  // Write integrated z back in place.
  for (int i = tid; i < 32 * Z_; i += 256) {
    int r = i >> 6, c = i & 63;
    int gr = blockIdx.x * 32 + r;
    int b = gr / (T_ - 1), t = gr % (T_ - 1);
    out[((size_t)b * T_ + t) * Z_ + c] = sm[ZB + r * 68 + c];
  }
}

extern "C" void kernel_launch(void* const* d_in, const int* in_sizes, int n_in,
                              void* d_out, int out_size, void* d_ws, size_t ws_size,
                              hipStream_t stream) {
  (void)in_sizes; (void)n_in; (void)out_size;
  const float* x    = (const float*)d_in[0];
  const float* h0   = (const float*)d_in[1];
  const float* Wih0 = (const float*)d_in[2];
  const float* Whh0 = (const float*)d_in[3];
  const float* bih0 = (const float*)d_in[4];
  const float* bhh0 = (const float*)d_in[5];
  const float* bih1 = (const float*)d_in[8];
  const float* Wih1 = (const float*)d_in[6];
  const float* Whh1 = (const float*)d_in[7];
  const float* bhh1 = (const float*)d_in[9];
  const float* Wh0  = (const float*)d_in[10];
  const float* bh0  = (const float*)d_in[11];
  const float* Wh1  = (const float*)d_in[12];
  const float* bh1  = (const float*)d_in[13];
  const float* Wout = (const float*)d_in[14];
  float* out = (float*)d_out;

  // Sequential RNN: single workgroup, LDS-resident weights/state (99,328 B).
  rnn_kernel<<<1, 256, 99328, stream>>>(x, h0, Wih0, Whh0, bih0, bhh0,
                                        Wih1, Whh1, bih1, bhh1, out);

  const size_t trBytes = (size_t)(H_ * H_ + Z_ * H_) * sizeof(float); // 327,680
  if (ws_size >= trBytes) {
    float* wsf = (float*)d_ws;
    transpose_kernel<<<(H_ * H_ + 255) / 256, 256, 0, stream>>>(Wh1, Wh0, wsf);
    ode_kernel<true><<<255, 256, 110080, stream>>>(
        Wh0, bh0, Wh1, bh1, Wout, wsf, wsf + H_ * H_, out);
  } else {
    ode_kernel<false><<<255, 256, 110080, stream>>>(
        Wh0, bh0, Wh1, bh1, Wout, Wh1, Wh0, out);
  }
}